// SymmetryAwareGeodesicLoss_61512521613368
// MI455X (gfx1250) — compile-verified
//
#include <hip/hip_runtime.h>
#include <hip/hip_bf16.h>
#include <math.h>

// -----------------------------------------------------------------------------
// SymmetryAwareGeodesicLoss for MI455X (gfx1250, wave32)
//
// Memory-bound streaming kernel: 36 B/element * 4.19M elements ~ 155 MB once
// through HBM -> ~6.7 us floor @ 23.3 TB/s. Main kernel uses float4 (B128)
// coalesced loads, one acosf per element, wave32 shuffle + LDS reduction.
// Final reduction of 2048 block partials is done in a single wave using
// V_WMMA_F32_16X16X4_F32 with an all-ones B matrix (collapses 32 lane values
// in one matrix op), which is the CDNA5 matrix path for f32.
// Deterministic: fixed grid, fixed reduction tree, no float atomics.
// -----------------------------------------------------------------------------

#define GEO_EPS 1e-6f

typedef __attribute__((ext_vector_type(2))) float v2f;
typedef __attribute__((ext_vector_type(8))) float v8f;

constexpr int BLOCK      = 256;   // 8 wave32 waves per block
constexpr int MAX_BLOCKS = 2048;  // 8 elements/thread at B = 4,194,304

__global__ __launch_bounds__(BLOCK) void geo_loss_partial(
    const float4* __restrict__ qpred,
    const float4* __restrict__ qgt,
    const int*    __restrict__ labels,
    float*        __restrict__ partials,
    int n)
{
    const int tid    = blockIdx.x * BLOCK + threadIdx.x;
    const int stride = gridDim.x * BLOCK;

    float acc = 0.0f;

    for (int i = tid; i < n; i += stride) {
        // quats stored (w, x, y, z) -> float4 (.x=w, .y=x, .z=y, .w=z)
        const float4 p = qpred[i];
        const float4 g = qgt[i];
        const int    lab = labels[i];

        // ---- normalize both quaternions (matches _normalize with eps) ----
        const float ssp = p.x*p.x + p.y*p.y + p.z*p.z + p.w*p.w;
        const float ssg = g.x*g.x + g.y*g.y + g.z*g.z + g.w*g.w;
        const float rp  = 1.0f / fmaxf(sqrtf(ssp), GEO_EPS);
        const float rg  = 1.0f / fmaxf(sqrtf(ssg), GEO_EPS);
        const float pw = p.x*rp, px = p.y*rp, py = p.z*rp, pz = p.w*rp;
        const float gw = g.x*rg, gx = g.y*rg, gy = g.z*rg, gz = g.w*rg;

        // ---- geodesic dots: identity sym and (0,0,0,1) sym ----
        // quat_mul(g, [0,0,0,1]) = (-gz, gy, -gx, gw)
        const float d1 = fabsf(pw*gw + px*gx + py*gy + pz*gz);
        const float d2 = fabsf(-pw*gz + px*gy - py*gx + pz*gw);

        const bool is_axial = (lab == 5)  || (lab == 11);
        const bool is_disc  = (lab == 10) || (lab == 12);

        // acos is monotone decreasing: min over angles == angle of max |dot|.
        float din = is_disc ? fmaxf(d1, d2) : d1;
        din = fminf(fmaxf(din, -1.0f + GEO_EPS), 1.0f - GEO_EPS);
        const float ang = 2.0f * acosf(din);

        // ---- rotate +Z by each quat: (2(wy+xz), 2(yz-wx), 1-2(x^2+y^2)) ----
        const float zpx = 2.0f * (pw*py + px*pz);
        const float zpy = 2.0f * (py*pz - pw*px);
        const float zpz = 1.0f - 2.0f * (px*px + py*py);
        const float zgx = 2.0f * (gw*gy + gx*gz);
        const float zgy = 2.0f * (gy*gz - gw*gx);
        const float zgz = 1.0f - 2.0f * (gx*gx + gy*gy);
        const float rzp = 1.0f / fmaxf(sqrtf(zpx*zpx + zpy*zpy + zpz*zpz), GEO_EPS);
        const float rzg = 1.0f / fmaxf(sqrtf(zgx*zgx + zgy*zgy + zgz*zgz), GEO_EPS);
        const float lax = 1.0f - (zpx*zgx + zpy*zgy + zpz*zgz) * (rzp * rzg);

        acc += is_axial ? lax : ang;
    }

    // ---- wave32 butterfly reduction ----
    #pragma unroll
    for (int m = 16; m > 0; m >>= 1) acc += __shfl_xor(acc, m, 32);

    __shared__ float smem[BLOCK / 32];
    const int lane = threadIdx.x & 31;
    const int wid  = threadIdx.x >> 5;
    if (lane == 0) smem[wid] = acc;
    __syncthreads();

    if (wid == 0) {
        float v = (lane < (BLOCK / 32)) ? smem[lane] : 0.0f;
        #pragma unroll
        for (int m = 16; m > 0; m >>= 1) v += __shfl_xor(v, m, 32);
        if (lane == 0) partials[blockIdx.x] = v;
    }
}

// Single-wave final reduction. Uses V_WMMA_F32_16X16X4_F32 with B = ones:
//   A[m][k] carries lane partials in K-slots 0 (lanes 0-15) and 2 (lanes 16-31),
//   so D[m][n] = s_m + s_{m+16} (row-sum r_m, replicated over columns).
//   Lanes 0-15 hold rows 0..7 in their 8 acc VGPRs, lanes 16-31 rows 8..15;
//   summing the 8 VGPRs then one shfl_xor(16) yields the full 32-lane sum.
__global__ __launch_bounds__(32) void geo_loss_final(
    const float* __restrict__ partials,
    float*       __restrict__ out,
    int nparts, float inv_n)
{
    const int lane = threadIdx.x;

    float s = 0.0f;
    for (int i = lane; i < nparts; i += 32) s += partials[i];
    // control flow reconverged here: EXEC all-1s for WMMA (single full wave)

    v2f a; a[0] = s;    a[1] = 0.0f;   // A matrix: per-lane partial, zero pad
    v2f b; b[0] = 1.0f; b[1] = 1.0f;   // B matrix: all ones (layout-invariant)
    v8f c = {};
    c = __builtin_amdgcn_wmma_f32_16x16x4_f32(
            /*neg_a=*/false, a, /*neg_b=*/false, b,
            /*c_mod=*/(short)0, c, /*reuse_a=*/false, /*reuse_b=*/false);

    float t = c[0] + c[1] + c[2] + c[3] + c[4] + c[5] + c[6] + c[7];
    t += __shfl_xor(t, 16, 32);

    if (lane == 0) out[0] = t * inv_n;
}

extern "C" void kernel_launch(void* const* d_in, const int* in_sizes, int n_in,
                              void* d_out, int out_size, void* d_ws, size_t ws_size,
                              hipStream_t stream)
{
    const float4* qpred  = (const float4*)d_in[0];
    const float4* qgt    = (const float4*)d_in[1];
    const int*    labels = (const int*)d_in[2];

    const int n = in_sizes[0] / 4;   // B quaternions

    float* partials = (float*)d_ws;
    int nblocks = MAX_BLOCKS;
    if (ws_size < (size_t)nblocks * sizeof(float)) {
        nblocks = (int)(ws_size / sizeof(float));
        if (nblocks < 1) nblocks = 1;
    }

    geo_loss_partial<<<nblocks, BLOCK, 0, stream>>>(qpred, qgt, labels, partials, n);
    geo_loss_final<<<1, 32, 0, stream>>>(partials, (float*)d_out, nblocks,
                                         1.0f / (float)n);
}